// CustomLoss_39436389712300
// MI455X (gfx1250) — compile-verified
//
#include <hip/hip_runtime.h>
#include <math.h>

// ---------------------------------------------------------------------------
// CustomLoss for MI455X (gfx1250, wave32).
//   Stage 1: histogram over (superpixel, class)       -- tiny, atomic ints
//   Stage 2: per-superpixel argmax (first-max ties)   -- tiny
//   Stage 3: fused single-pass online log-sum-exp + gather + refs + mask.
//            - mode table DMA'd to LDS via TENSOR_LOAD_TO_LDS (TDM) + s_wait_tensorcnt
//            - 4 pixels/thread => global_load_b128 class slices (HBM-bound: B*C*N*4 B vs 23.3 TB/s)
//            - loss reduced per-wave on the f32 matrix pipe (V_WMMA_F32_16X16X4_F32, ones-B)
//   Stage 4: finalize mean.
// NOTE: B (batch) not derivable from in_sizes; histogram keys by superpixel
// only (B==1 assumption). Loss math is batch-size-agnostic (global mean).
// ---------------------------------------------------------------------------

#define N_SP 1024

typedef __attribute__((ext_vector_type(2))) float        v2f;
typedef __attribute__((ext_vector_type(8))) float        v8f;
typedef __attribute__((ext_vector_type(4))) unsigned int u32x4;
typedef __attribute__((ext_vector_type(8))) int          i32x8;
typedef __attribute__((ext_vector_type(4))) int          i32x4;

// Zero hist (N_SP*C ints), loss accumulator (1 float), mask region of d_out (C floats).
__global__ void zero_kernel(int* __restrict__ hist, float* __restrict__ loss_acc,
                            float* __restrict__ mask, int nhist, int C) {
  int tid = blockIdx.x * blockDim.x + threadIdx.x;
  int stride = gridDim.x * blockDim.x;
  for (int i = tid; i < nhist; i += stride) hist[i] = 0;
  if (tid < C) mask[tid] = 0.0f;
  if (tid == 0) *loss_acc = 0.0f;
}

__global__ void hist_kernel(const int* __restrict__ tgt, const int* __restrict__ sp,
                            int* __restrict__ hist, int BN, int C) {
  int tid = blockIdx.x * blockDim.x + threadIdx.x;
  int stride = gridDim.x * blockDim.x;
  for (int i = tid; i < BN; i += stride) {
    atomicAdd(&hist[sp[i] * C + tgt[i]], 1);
  }
}

// argmax over classes, ties -> smallest class id (first max), matches np.argmax.
__global__ void mode_kernel(const int* __restrict__ hist, int* __restrict__ mode, int C) {
  int spi = blockIdx.x * blockDim.x + threadIdx.x;
  if (spi >= N_SP) return;
  const int* h = hist + (size_t)spi * C;
  int best = h[0], bi = 0;
  for (int c = 1; c < C; ++c) {
    int v = h[c];
    if (v > best) { best = v; bi = c; }
  }
  mode[spi] = bi;
}

// Online log-sum-exp update (rescale branch is rarely taken after warmup).
__device__ __forceinline__ void lse_step(float x, float& m, float& s) {
  if (x > m) { s = s * __expf(m - x) + 1.0f; m = x; }
  else       { s += __expf(x - m); }
}

__global__ void ce_main_kernel(const float* __restrict__ logits,
                               const int* __restrict__ sp,
                               const int* __restrict__ gmode,
                               float* __restrict__ dout,      // [0]=loss, [1..BN]=refs, [1+BN..]=mask
                               float* __restrict__ loss_acc,
                               int BN, int C, int nv4) {
  __shared__ int smode[N_SP];

  // ---- TDM: DMA the 4KB mode table global->LDS (wave 0 issues; per-wave op) ----
  if (threadIdx.x < 32) {
    unsigned lds = (unsigned)(size_t)(void*)smode;                 // LDS byte offset (low 32 bits)
    unsigned long long ga = (unsigned long long)(size_t)gmode;
    u32x4 g0;
    g0[0] = 1u;                                                    // count=1 valid, no gather
    g0[1] = lds;                                                   // lds_addr [63:32]
    g0[2] = (unsigned)(ga & 0xFFFFFFFFu);                          // global_addr[31:0]
    g0[3] = (unsigned)((ga >> 32) & 0x01FFFFFFu) | 0x80000000u;    // ga[56:32] | type=2
    i32x8 g1 = {};
    g1[0] = 2 << 16;                                               // data_size = 4 bytes
    g1[1] = (int)((N_SP & 0xFFFFu) << 16);                         // tensor_dim0[15:0] = 1024
    g1[2] = (int)((N_SP >> 16) | (1u << 16));                      // tensor_dim0[31:16], tensor_dim1 = 1
    g1[3] = (int)((N_SP & 0xFFFFu) << 16);                         // tile_dim0 = 1024
    g1[4] = 1;                                                     // tile_dim1 = 1
    g1[5] = N_SP;                                                  // tensor_dim0_stride[31:0]
    g1[6] = (int)((N_SP & 0xFFFFu) << 16);                         // tensor_dim1_stride[15:0]
    g1[7] = 0;
    i32x4 gz = {};
#if __clang_major__ >= 23
    i32x8 gz8 = {};
    __builtin_amdgcn_tensor_load_to_lds(g0, g1, gz, gz, gz8, 0);
#else
    __builtin_amdgcn_tensor_load_to_lds(g0, g1, gz, gz, 0);
#endif
    __builtin_amdgcn_s_wait_tensorcnt(0);                          // DMA complete before barrier
  }
  __syncthreads();

  int tid = blockIdx.x * blockDim.x + threadIdx.x;
  int stride = gridDim.x * blockDim.x;
  float p = 0.0f;  // per-lane NLL partial

  // ---- vector path: 4 pixels/thread, b128 class-slice loads (BN % 4 == 0) ----
  for (int i = tid; i < nv4; i += stride) {
    int4 s4 = ((const int4*)sp)[i];
    int r0 = smode[s4.x], r1 = smode[s4.y], r2 = smode[s4.z], r3 = smode[s4.w];
    float m0 = -INFINITY, m1 = m0, m2 = m0, m3 = m0;
    float s0 = 0.f, s1 = 0.f, s2 = 0.f, s3 = 0.f;
    float x0 = 0.f, x1 = 0.f, x2 = 0.f, x3 = 0.f;
    for (int c = 0; c < C; ++c) {
      float4 x = ((const float4*)(logits + (size_t)c * BN))[i];
      if (c + 1 < C)
        __builtin_prefetch(logits + (size_t)(c + 1) * BN + 4 * (size_t)i, 0, 0);
      if (c == r0) x0 = x.x;
      if (c == r1) x1 = x.y;
      if (c == r2) x2 = x.z;
      if (c == r3) x3 = x.w;
      lse_step(x.x, m0, s0);
      lse_step(x.y, m1, s1);
      lse_step(x.z, m2, s2);
      lse_step(x.w, m3, s3);
    }
    p += (__logf(s0) + m0 - x0) + (__logf(s1) + m1 - x1)
       + (__logf(s2) + m2 - x2) + (__logf(s3) + m3 - x3);
    int base = 1 + 4 * i;
    dout[base + 0] = (float)r0;
    dout[base + 1] = (float)r1;
    dout[base + 2] = (float)r2;
    dout[base + 3] = (float)r3;
    dout[1 + BN + r0] = 1.0f;  // presence mask (benign same-value races)
    dout[1 + BN + r1] = 1.0f;
    dout[1 + BN + r2] = 1.0f;
    dout[1 + BN + r3] = 1.0f;
  }

  // ---- scalar tail (also the whole range when BN % 4 != 0) ----
  for (int i = 4 * nv4 + tid; i < BN; i += stride) {
    int r = smode[sp[i]];
    float m = -INFINITY, s = 0.0f, xr = 0.0f;
    for (int c = 0; c < C; ++c) {
      float x = logits[(size_t)c * BN + i];
      if (c == r) xr = x;
      lse_step(x, m, s);
    }
    p += __logf(s) + m - xr;
    dout[1 + i] = (float)r;
    dout[1 + BN + r] = 1.0f;
  }

  // ---- wave32 reduction on the matrix pipe: D = A x ones(4x16) + 0 ----
  // A (16x4 f32): lanes 0-15 carry {K0=p, K1=0}, lanes 16-31 {K2=p, K3=0}.
  // Row m of D == p[m] + p[m+16]; lanes 0-15 see rows 0-7 in D[0..7],
  // lanes 16-31 see rows 8-15. EXEC is all-ones (no early returns above).
  v2f a; a.x = p;    a.y = 0.0f;
  v2f b; b.x = 1.0f; b.y = 1.0f;   // all-ones B covers any 4x16 layout
  v8f acc = {};
  acc = __builtin_amdgcn_wmma_f32_16x16x4_f32(
      /*neg_a=*/false, a, /*neg_b=*/false, b,
      /*c_mod=*/(short)0, acc, /*reuse_a=*/false, /*reuse_b=*/false);
  float s8 = acc[0] + acc[1] + acc[2] + acc[3] + acc[4] + acc[5] + acc[6] + acc[7];
  float tot = s8 + __shfl_down(s8, 16);      // rows 0-7 (+) rows 8-15
  if ((threadIdx.x & 31) == 0) atomicAdd(loss_acc, tot);
}

__global__ void finalize_kernel(const float* __restrict__ loss_acc,
                                float* __restrict__ dout, int BN) {
  if (blockIdx.x == 0 && threadIdx.x == 0)
    dout[0] = *loss_acc / (float)BN;
}

extern "C" void kernel_launch(void* const* d_in, const int* in_sizes, int n_in,
                              void* d_out, int out_size, void* d_ws, size_t ws_size,
                              hipStream_t stream) {
  const float* logits = (const float*)d_in[0];  // [B, C, H, W] flat
  const int*   tgt    = (const int*)d_in[1];    // [B, H, W] flat
  const int*   sp     = (const int*)d_in[2];    // [B, H, W] flat

  const int BN = in_sizes[1];                   // B*H*W
  const int C  = in_sizes[0] / in_sizes[1];     // channels

  float* dout = (float*)d_out;

  // Workspace layout
  float* loss_acc = (float*)d_ws;                        // 1 float (padded to 64B)
  int*   hist     = (int*)d_ws + 16;                     // N_SP*C ints
  int*   mode     = hist + (size_t)N_SP * C;             // N_SP ints
  const int nhist = N_SP * C;

  const int TB = 256;  // 8 wave32s per block
  int zb = (nhist + TB - 1) / TB; if (zb < 1) zb = 1;
  zero_kernel<<<zb, TB, 0, stream>>>(hist, loss_acc, dout + 1 + BN, nhist, C);

  int hb = (BN + TB - 1) / TB; if (hb > 4096) hb = 4096;
  hist_kernel<<<hb, TB, 0, stream>>>(tgt, sp, hist, BN, C);

  mode_kernel<<<(N_SP + TB - 1) / TB, TB, 0, stream>>>(hist, mode, C);

  const int nv4 = ((BN & 3) == 0) ? (BN >> 2) : 0;       // vector path needs 16B slice alignment
  int work = (nv4 > 0) ? nv4 : BN;
  int mb = (work + TB - 1) / TB; if (mb > 4096) mb = 4096; if (mb < 1) mb = 1;
  ce_main_kernel<<<mb, TB, 0, stream>>>(logits, sp, mode, dout, loss_acc, BN, C, nv4);

  finalize_kernel<<<1, 32, 0, stream>>>(loss_acc, dout, BN);
}